// MultiHeadGENLayer_34299608826152
// MI455X (gfx1250) — compile-verified
//
#include <hip/hip_runtime.h>
#include <hip/hip_bf16.h>
#include <math.h>

#define N_NODES 50000
#define N_EDGES 800000
#define DIM     128
#define DIM2    256

typedef __attribute__((ext_vector_type(2))) float v2f;
typedef __attribute__((ext_vector_type(8))) float v8f;

__device__ __forceinline__ v8f wmma4(v2f a, v2f b, v8f c) {
  // D = A(16x4 f32) * B(4x16 f32) + C(16x16 f32)
  return __builtin_amdgcn_wmma_f32_16x16x4_f32(
      /*neg_a=*/false, a, /*neg_b=*/false, b,
      /*c_mod=*/(short)0, c, /*reuse_a=*/false, /*reuse_b=*/false);
}

__device__ __forceinline__ void atomicMaxFloat(float* addr, float val) {
  if (val >= 0.0f) atomicMax((int*)addr, __float_as_int(val));
  else             atomicMin((unsigned int*)addr, __float_as_uint(val));
}

// ---------------------------------------------------------------------------
// 1) workspace init
// ---------------------------------------------------------------------------
__global__ __launch_bounds__(256) void init_ws(float* __restrict__ smax,
                                               float* __restrict__ denom,
                                               float* __restrict__ aggr,
                                               float* __restrict__ bnsum,
                                               float* __restrict__ bnsumsq) {
  int i = blockIdx.x * 256 + threadIdx.x;           // grid covers N*DIM exactly
  smax[i]  = -INFINITY;
  denom[i] = 0.0f;
  aggr[i]  = 0.0f;
  if (i < DIM2) { bnsum[i] = 0.0f; bnsumsq[i] = 0.0f; }
}

// ---------------------------------------------------------------------------
// 2-4) edge scatter-softmax passes (1 wave per edge, 4 channels per lane)
// ---------------------------------------------------------------------------
__device__ __forceinline__ void edge_msg(const float* __restrict__ x,
                                         const float* __restrict__ wedge,
                                         int s, float a, int c, float* m) {
  float4 xv = *(const float4*)(x + (size_t)s * DIM + c);
  float4 wv = *(const float4*)(wedge + c);
  const float* xp = (const float*)&xv;
  const float* wp = (const float*)&wv;
#pragma unroll
  for (int j = 0; j < 4; ++j)
    m[j] = fmaxf(xp[j] + a * wp[j], 0.0f) + 1e-7f;
}

__global__ __launch_bounds__(256) void edge_pass_max(
    const float* __restrict__ x, const int* __restrict__ src,
    const int* __restrict__ dst, const float* __restrict__ ea,
    const float* __restrict__ wedge, const float* __restrict__ tptr,
    float* __restrict__ smax) {
  int e = blockIdx.x * 8 + (threadIdx.x >> 5);
  int lane = threadIdx.x & 31;
  int s = src[e], d = dst[e];
  float a = ea[e], t = tptr[0];
  int c = lane * 4;
  float m[4];
  edge_msg(x, wedge, s, a, c, m);
  float* base = smax + (size_t)d * DIM + c;
#pragma unroll
  for (int j = 0; j < 4; ++j) atomicMaxFloat(base + j, m[j] * t);
}

__global__ __launch_bounds__(256) void edge_pass_denom(
    const float* __restrict__ x, const int* __restrict__ src,
    const int* __restrict__ dst, const float* __restrict__ ea,
    const float* __restrict__ wedge, const float* __restrict__ tptr,
    const float* __restrict__ smax, float* __restrict__ denom) {
  int e = blockIdx.x * 8 + (threadIdx.x >> 5);
  int lane = threadIdx.x & 31;
  int s = src[e], d = dst[e];
  float a = ea[e], t = tptr[0];
  int c = lane * 4;
  float m[4];
  edge_msg(x, wedge, s, a, c, m);
  float4 sm = *(const float4*)(smax + (size_t)d * DIM + c);
  const float* sp = (const float*)&sm;
  float* base = denom + (size_t)d * DIM + c;
#pragma unroll
  for (int j = 0; j < 4; ++j) atomicAdd(base + j, expf(m[j] * t - sp[j]));
}

__global__ __launch_bounds__(256) void edge_pass_aggr(
    const float* __restrict__ x, const int* __restrict__ src,
    const int* __restrict__ dst, const float* __restrict__ ea,
    const float* __restrict__ wedge, const float* __restrict__ tptr,
    const float* __restrict__ smax, const float* __restrict__ denom,
    float* __restrict__ aggr) {
  int e = blockIdx.x * 8 + (threadIdx.x >> 5);
  int lane = threadIdx.x & 31;
  int s = src[e], d = dst[e];
  float a = ea[e], t = tptr[0];
  int c = lane * 4;
  float m[4];
  edge_msg(x, wedge, s, a, c, m);
  float4 sm = *(const float4*)(smax + (size_t)d * DIM + c);
  float4 dn = *(const float4*)(denom + (size_t)d * DIM + c);
  const float* sp = (const float*)&sm;
  const float* dp = (const float*)&dn;
  float* base = aggr + (size_t)d * DIM + c;
#pragma unroll
  for (int j = 0; j < 4; ++j) {
    float ex = expf(m[j] * t - sp[j]);
    atomicAdd(base + j, m[j] * (ex / (dp[j] + 1e-16f)));
  }
}

// ---------------------------------------------------------------------------
// 5) h1 = (aggr + x) @ W1^T   [N x 256], WMMA f32 16x16x4, K = 128
//    512 threads = 16 waves; wave w -> output cols [16w, 16w+16)
// ---------------------------------------------------------------------------
__global__ __launch_bounds__(512) void gemm_node1(
    const float* __restrict__ aggr, const float* __restrict__ x,
    const float* __restrict__ W1, float* __restrict__ h1) {
  __shared__ float As[16 * DIM];
  int m0 = blockIdx.x * 16;
  int t = threadIdx.x;
  {
    int idx = t * 4;                       // 512*4 = 2048 = 16*128
    int r = idx >> 7, c = idx & 127;
    float4 av = *(const float4*)(aggr + (size_t)(m0 + r) * DIM + c);
    float4 xv = *(const float4*)(x + (size_t)(m0 + r) * DIM + c);
    float4 hv;
    hv.x = av.x + xv.x; hv.y = av.y + xv.y;
    hv.z = av.z + xv.z; hv.w = av.w + xv.w;
    *(float4*)(As + idx) = hv;
  }
  __syncthreads();

  int wave = t >> 5, lane = t & 31;
  int n0 = wave * 16;
  int col = lane & 15, half = lane >> 4;
  const float* Arow = As + col * DIM;
  const float* Brow = W1 + (size_t)(n0 + col) * DIM;   // B[k][n] = W1[n][k]
  v8f acc = {};
#pragma unroll
  for (int k = 0; k < DIM; k += 4) {
    int ko = k + 2 * half;
    v2f a; a.x = Arow[ko]; a.y = Arow[ko + 1];
    v2f b; b.x = Brow[ko]; b.y = Brow[ko + 1];
    acc = wmma4(a, b, acc);
  }
#pragma unroll
  for (int v = 0; v < 8; ++v)
    h1[(size_t)(m0 + v + 8 * half) * DIM2 + n0 + col] = acc[v];
}

// ---------------------------------------------------------------------------
// 6) BatchNorm batch statistics: per-column sum / sumsq over 50000 rows
// ---------------------------------------------------------------------------
__global__ __launch_bounds__(256) void bn_stats(const float* __restrict__ h1,
                                                float* __restrict__ bnsum,
                                                float* __restrict__ bnsumsq) {
  int col = threadIdx.x;
  int rpb = (N_NODES + gridDim.x - 1) / gridDim.x;
  int r0 = blockIdx.x * rpb;
  int r1 = min(r0 + rpb, N_NODES);
  float s = 0.0f, ss = 0.0f;
  for (int r = r0; r < r1; ++r) {
    float v = h1[(size_t)r * DIM2 + col];
    s += v; ss += v * v;
  }
  atomicAdd(bnsum + col, s);
  atomicAdd(bnsumsq + col, ss);
}

// ---------------------------------------------------------------------------
// 7) out_conv = relu(bn(h1)) @ W2^T  [N x 128], K = 256 (BN fused into A load)
// ---------------------------------------------------------------------------
__global__ __launch_bounds__(256) void gemm_node2(
    const float* __restrict__ h1, const float* __restrict__ bnsum,
    const float* __restrict__ bnsumsq, const float* __restrict__ gamma,
    const float* __restrict__ beta, const float* __restrict__ W2,
    float* __restrict__ outx) {
  __shared__ float As[16 * DIM2];
  int m0 = blockIdx.x * 16;
  int t = threadIdx.x;
  const float invN = 1.0f / (float)N_NODES;
#pragma unroll
  for (int i = 0; i < 4; ++i) {
    int idx = t * 4 + i * 1024;            // 4*1024 = 4096 = 16*256
    int r = idx >> 8, c = idx & 255;
    float4 hv = *(const float4*)(h1 + (size_t)(m0 + r) * DIM2 + c);
    const float* hp = (const float*)&hv;
#pragma unroll
    for (int j = 0; j < 4; ++j) {
      int cc = c + j;
      float mean = bnsum[cc] * invN;
      float var  = bnsumsq[cc] * invN - mean * mean;
      float v = (hp[j] - mean) * rsqrtf(var + 1e-5f) * gamma[cc] + beta[cc];
      As[idx + j] = fmaxf(v, 0.0f);
    }
  }
  __syncthreads();

  int wave = t >> 5, lane = t & 31;
  int n0 = wave * 16;                       // 8 waves cover 128 output cols
  int col = lane & 15, half = lane >> 4;
  const float* Arow = As + col * DIM2;
  const float* Brow = W2 + (size_t)(n0 + col) * DIM2;  // B[k][n] = W2[n][k]
  v8f acc = {};
#pragma unroll
  for (int k = 0; k < DIM2; k += 4) {
    int ko = k + 2 * half;
    v2f a; a.x = Arow[ko]; a.y = Arow[ko + 1];
    v2f b; b.x = Brow[ko]; b.y = Brow[ko + 1];
    acc = wmma4(a, b, acc);
  }
#pragma unroll
  for (int v = 0; v < 8; ++v)
    outx[(size_t)(m0 + v + 8 * half) * DIM + n0 + col] = acc[v];
}

// ---------------------------------------------------------------------------
// 8) in-place LayerNorm + ELU per node row (1 wave per row, wave32 shuffles)
// ---------------------------------------------------------------------------
__global__ __launch_bounds__(256) void ln_elu_rows(float* __restrict__ data,
                                                   const float* __restrict__ g,
                                                   const float* __restrict__ b,
                                                   int do_elu) {
  int row = blockIdx.x * 8 + (threadIdx.x >> 5);
  int lane = threadIdx.x & 31;
  float* p = data + (size_t)row * DIM + lane * 4;
  float4 v = *(float4*)p;
  float* vp = (float*)&v;
  float s = vp[0] + vp[1] + vp[2] + vp[3];
  float ss = vp[0]*vp[0] + vp[1]*vp[1] + vp[2]*vp[2] + vp[3]*vp[3];
#pragma unroll
  for (int o = 16; o > 0; o >>= 1) {
    s  += __shfl_xor(s, o, 32);
    ss += __shfl_xor(ss, o, 32);
  }
  float mean = s * (1.0f / 128.0f);
  float var  = ss * (1.0f / 128.0f) - mean * mean;
  float inv  = rsqrtf(var + 1e-5f);
#pragma unroll
  for (int j = 0; j < 4; ++j) {
    int c = lane * 4 + j;
    float y = (vp[j] - mean) * inv * g[c] + b[c];
    if (do_elu) y = (y > 0.0f) ? y : expm1f(y);
    vp[j] = y;
  }
  *(float4*)p = v;
}

// ---------------------------------------------------------------------------
// 9) out_e = gelu(concat(out_x[src], out_x[dst]) @ We^T + be)  [E x 128]
//    gather-concat A tile staged in LDS; K = 256; fused bias + exact GELU
// ---------------------------------------------------------------------------
__global__ __launch_bounds__(256) void gemm_edge(
    const float* __restrict__ outx, const int* __restrict__ src,
    const int* __restrict__ dst, const float* __restrict__ We,
    const float* __restrict__ be, float* __restrict__ oute) {
  __shared__ float As[16 * DIM2];
  int m0 = blockIdx.x * 16;
  int t = threadIdx.x;
  {
    int idx = t * 16;                      // 256*16 = 4096 = 16*256
    int r = idx >> 8, c = idx & 255;
    int node = (c < DIM) ? src[m0 + r] : dst[m0 + r];
    const float* base = outx + (size_t)node * DIM + (c & 127);
#pragma unroll
    for (int i = 0; i < 4; ++i)
      *(float4*)(As + idx + i * 4) = *(const float4*)(base + i * 4);
  }
  __syncthreads();

  int wave = t >> 5, lane = t & 31;
  int n0 = wave * 16;
  int col = lane & 15, half = lane >> 4;
  const float* Arow = As + col * DIM2;
  const float* Brow = We + (size_t)(n0 + col) * DIM2;  // B[k][n] = We[n][k]
  v8f acc = {};
#pragma unroll
  for (int k = 0; k < DIM2; k += 4) {
    int ko = k + 2 * half;
    v2f a; a.x = Arow[ko]; a.y = Arow[ko + 1];
    v2f b; b.x = Brow[ko]; b.y = Brow[ko + 1];
    acc = wmma4(a, b, acc);
  }
  int n = n0 + col;
  float bias = be[n];
#pragma unroll
  for (int v = 0; v < 8; ++v) {
    float z = acc[v] + bias;
    float gz = 0.5f * z * (1.0f + erff(z * 0.70710678118654752f));
    oute[(size_t)(m0 + v + 8 * half) * DIM + n] = gz;
  }
}

// ---------------------------------------------------------------------------
// launcher
// ---------------------------------------------------------------------------
extern "C" void kernel_launch(void* const* d_in, const int* in_sizes, int n_in,
                              void* d_out, int out_size, void* d_ws, size_t ws_size,
                              hipStream_t stream) {
  const float* x      = (const float*)d_in[0];
  const int*   ei     = (const int*)d_in[1];
  const float* ea     = (const float*)d_in[2];
  const float* wedge  = (const float*)d_in[3];
  const float* tptr   = (const float*)d_in[4];
  const float* W1     = (const float*)d_in[5];
  const float* bn_g   = (const float*)d_in[6];
  const float* bn_b   = (const float*)d_in[7];
  const float* W2     = (const float*)d_in[8];
  const float* ln_g   = (const float*)d_in[9];
  const float* ln_b   = (const float*)d_in[10];
  const float* We     = (const float*)d_in[11];
  const float* be     = (const float*)d_in[12];
  const float* lne_g  = (const float*)d_in[13];
  const float* lne_b  = (const float*)d_in[14];

  const int* src = ei;                 // edge_index[0, :]
  const int* dst = ei + N_EDGES;       // edge_index[1, :]

  float* outx = (float*)d_out;                       // [N, 128]
  float* oute = outx + (size_t)N_NODES * DIM;        // [E, 128]

  float* ws      = (float*)d_ws;
  float* smax    = ws;                                    // N*128
  float* denom   = smax  + (size_t)N_NODES * DIM;         // N*128
  float* aggr    = denom + (size_t)N_NODES * DIM;         // N*128
  float* h1      = aggr  + (size_t)N_NODES * DIM;         // N*256
  float* bnsum   = h1    + (size_t)N_NODES * DIM2;        // 256
  float* bnsumsq = bnsum + DIM2;                          // 256

  // 1) init (N*DIM = 6,400,000 = 25000 * 256 exactly)
  init_ws<<<25000, 256, 0, stream>>>(smax, denom, aggr, bnsum, bnsumsq);

  // 2-4) scatter softmax over edges (E = 800000 = 100000 blocks * 8 waves)
  edge_pass_max  <<<100000, 256, 0, stream>>>(x, src, dst, ea, wedge, tptr, smax);
  edge_pass_denom<<<100000, 256, 0, stream>>>(x, src, dst, ea, wedge, tptr, smax, denom);
  edge_pass_aggr <<<100000, 256, 0, stream>>>(x, src, dst, ea, wedge, tptr, smax, denom, aggr);

  // 5) h1 = (aggr + x) @ W1^T      (N/16 = 3125 row tiles, exact)
  gemm_node1<<<3125, 512, 0, stream>>>(aggr, x, W1, h1);

  // 6) BN batch stats
  bn_stats<<<256, 256, 0, stream>>>(h1, bnsum, bnsumsq);

  // 7) out_conv = relu(bn(h1)) @ W2^T -> d_out node region
  gemm_node2<<<3125, 256, 0, stream>>>(h1, bnsum, bnsumsq, bn_g, bn_b, W2, outx);

  // 8) out_x = elu(layer_norm(out_conv)) in-place (N/8 = 6250)
  ln_elu_rows<<<6250, 256, 0, stream>>>(outx, ln_g, ln_b, 1);

  // 9) edge MLP GEMM + bias + GELU -> d_out edge region (E/16 = 50000)
  gemm_edge<<<50000, 256, 0, stream>>>(outx, src, dst, We, be, oute);

  // 10) out_e = layer_norm(...) in-place (E/8 = 100000)
  ln_elu_rows<<<100000, 256, 0, stream>>>(oute, lne_g, lne_b, 0);
}